// Model_53901839564895
// MI455X (gfx1250) — compile-verified
//
#include <hip/hip_runtime.h>
#include <hip/hip_bf16.h>
#include <math.h>

typedef __attribute__((ext_vector_type(2))) float v2f;
typedef __attribute__((ext_vector_type(8))) float v8f;

// ---------------------------------------------------------------------------
// K1: conv 5x5 (3->5, pad 1) + ReLU + BN(eval) + maxpool 3/2/pad1
// x[512,3,128,128] -> p1[512,5,63,63]
// Phase 2 register-blocks 3 adjacent conv columns per work item so each LDS
// operand feeds ~2 FMAs and each weight load feeds 3 FMAs (LDS-bw relief).
// ---------------------------------------------------------------------------
__global__ __launch_bounds__(256) void conv1_bn_pool(
    const float* __restrict__ x, const float* __restrict__ k1,
    const float* __restrict__ gamma, const float* __restrict__ beta,
    float* __restrict__ p1)
{
    __shared__ float sIn[3 * 37 * 37];    // input tile   (16.4 KB)
    __shared__ float sConv[5 * 33 * 33];  // conv+bn tile (21.8 KB)

    const int b   = blockIdx.y;
    const int pi0 = (blockIdx.x >> 2) * 16;
    const int pj0 = (blockIdx.x & 3) * 16;
    const int cy0 = 2 * pi0 - 1;
    const int cx0 = 2 * pj0 - 1;
    const int tid = threadIdx.x;

    // Phase 1: input tile (zero outside [0,128))
    for (int idx = tid; idx < 3 * 37 * 37; idx += 256) {
        int c   = idx / (37 * 37);
        int rem = idx - c * 37 * 37;
        int r   = rem / 37, col = rem - r * 37;
        int iy  = cy0 - 1 + r, ix = cx0 - 1 + col;
        float v = 0.0f;
        if ((unsigned)iy < 128u && (unsigned)ix < 128u)
            v = x[((b * 3 + c) * 128 + iy) * 128 + ix];
        sIn[idx] = v;
    }
    __syncthreads();

    // Phase 2: conv+relu+bn, 3 columns per item. items = 5 * 33 * 11 = 1815
    const float rsq = 1.0f / sqrtf(1.0f + 1e-5f);
    for (int item = tid; item < 5 * 33 * 11; item += 256) {
        int o   = item / (33 * 11);
        int rem = item - o * (33 * 11);
        int r   = rem / 11, g = rem - r * 11;     // g: column group (3 cols)
        int c0  = 3 * g;
        float a0 = 0.f, a1 = 0.f, a2 = 0.f;
        #pragma unroll
        for (int c = 0; c < 3; ++c) {
            #pragma unroll
            for (int ki = 0; ki < 5; ++ki) {
                const float* rp = &sIn[c * 1369 + (r + ki) * 37 + c0];
                float i0 = rp[0], i1 = rp[1], i2 = rp[2], i3 = rp[3];
                float i4 = rp[4], i5 = rp[5], i6 = rp[6];
                const float* wp = &k1[(c * 25 + ki * 5) * 5 + o];
                float w0 = wp[0], w1 = wp[5], w2 = wp[10], w3 = wp[15], w4 = wp[20];
                a0 = fmaf(i0, w0, a0); a1 = fmaf(i1, w0, a1); a2 = fmaf(i2, w0, a2);
                a0 = fmaf(i1, w1, a0); a1 = fmaf(i2, w1, a1); a2 = fmaf(i3, w1, a2);
                a0 = fmaf(i2, w2, a0); a1 = fmaf(i3, w2, a1); a2 = fmaf(i4, w2, a2);
                a0 = fmaf(i3, w3, a0); a1 = fmaf(i4, w3, a1); a2 = fmaf(i5, w3, a2);
                a0 = fmaf(i4, w4, a0); a1 = fmaf(i5, w4, a1); a2 = fmaf(i6, w4, a2);
            }
        }
        const int cy = cy0 + r;
        const float gm = gamma[o], bt = beta[o];
        float acc[3] = {a0, a1, a2};
        #pragma unroll
        for (int j = 0; j < 3; ++j) {
            int cx = cx0 + c0 + j;
            float v = 0.0f;
            if ((unsigned)cy < 126u && (unsigned)cx < 126u)
                v = gm * (fmaxf(acc[j], 0.0f) * rsq) + bt;
            sConv[o * 1089 + r * 33 + c0 + j] = v;
        }
    }
    __syncthreads();

    // Phase 3: 3x3 stride-2 max pool (zero padding encoded in sConv)
    for (int idx = tid; idx < 5 * 16 * 16; idx += 256) {
        int o   = idx >> 8;
        int rem = idx & 255;
        int pr  = rem >> 4, pc = rem & 15;
        int py  = pi0 + pr, px = pj0 + pc;
        if (py < 63 && px < 63) {
            float m = -INFINITY;
            #pragma unroll
            for (int dy = 0; dy < 3; ++dy)
                #pragma unroll
                for (int dx = 0; dx < 3; ++dx)
                    m = fmaxf(m, sConv[o * 1089 + (2 * pr + dy) * 33 + (2 * pc + dx)]);
            p1[((b * 5 + o) * 63 + py) * 63 + px] = m;
        }
    }
}

// ---------------------------------------------------------------------------
// K2: conv 3x3 (5->9, pad 1) + ReLU + maxpool 3/2/pad1
// p1[512,5,63,63] -> p2[512,9,32,32]
// ---------------------------------------------------------------------------
__global__ __launch_bounds__(256) void conv2_pool(
    const float* __restrict__ p1, const float* __restrict__ k2,
    float* __restrict__ p2)
{
    __shared__ float sIn[5 * 35 * 35];    // 24.5 KB
    __shared__ float sConv[9 * 33 * 33];  // 39.2 KB (total 63.7 KB)

    const int b   = blockIdx.y;
    const int pi0 = (blockIdx.x >> 1) * 16;
    const int pj0 = (blockIdx.x & 1) * 16;
    const int cy0 = 2 * pi0 - 1;
    const int cx0 = 2 * pj0 - 1;
    const int tid = threadIdx.x;

    for (int idx = tid; idx < 5 * 35 * 35; idx += 256) {
        int c   = idx / (35 * 35);
        int rem = idx - c * 35 * 35;
        int r   = rem / 35, col = rem - r * 35;
        int iy  = cy0 - 1 + r, ix = cx0 - 1 + col;
        float v = 0.0f;
        if ((unsigned)iy < 63u && (unsigned)ix < 63u)
            v = p1[((b * 5 + c) * 63 + iy) * 63 + ix];
        sIn[idx] = v;
    }
    __syncthreads();

    // conv+relu, 3 columns per item. items = 9 * 33 * 11 = 3267
    for (int item = tid; item < 9 * 33 * 11; item += 256) {
        int o   = item / (33 * 11);
        int rem = item - o * (33 * 11);
        int r   = rem / 11, g = rem - r * 11;
        int c0  = 3 * g;
        float a0 = 0.f, a1 = 0.f, a2 = 0.f;
        #pragma unroll
        for (int c = 0; c < 5; ++c) {
            #pragma unroll
            for (int ki = 0; ki < 3; ++ki) {
                const float* rp = &sIn[c * 1225 + (r + ki) * 35 + c0];
                float i0 = rp[0], i1 = rp[1], i2 = rp[2], i3 = rp[3], i4 = rp[4];
                const float* wp = &k2[(c * 9 + ki * 3) * 9 + o];
                float w0 = wp[0], w1 = wp[9], w2 = wp[18];
                a0 = fmaf(i0, w0, a0); a1 = fmaf(i1, w0, a1); a2 = fmaf(i2, w0, a2);
                a0 = fmaf(i1, w1, a0); a1 = fmaf(i2, w1, a1); a2 = fmaf(i3, w1, a2);
                a0 = fmaf(i2, w2, a0); a1 = fmaf(i3, w2, a1); a2 = fmaf(i4, w2, a2);
            }
        }
        const int cy = cy0 + r;
        float acc[3] = {a0, a1, a2};
        #pragma unroll
        for (int j = 0; j < 3; ++j) {
            int cx = cx0 + c0 + j;
            float v = 0.0f;
            if ((unsigned)cy < 63u && (unsigned)cx < 63u)
                v = fmaxf(acc[j], 0.0f);
            sConv[o * 1089 + r * 33 + c0 + j] = v;
        }
    }
    __syncthreads();

    for (int idx = tid; idx < 9 * 16 * 16; idx += 256) {
        int o   = idx >> 8;
        int rem = idx & 255;
        int pr  = rem >> 4, pc = rem & 15;
        int py  = pi0 + pr, px = pj0 + pc;   // always < 32
        float m = -INFINITY;
        #pragma unroll
        for (int dy = 0; dy < 3; ++dy)
            #pragma unroll
            for (int dx = 0; dx < 3; ++dx)
                m = fmaxf(m, sConv[o * 1089 + (2 * pr + dy) * 33 + (2 * pc + dx)]);
        p2[((b * 9 + o) * 32 + py) * 32 + px] = m;
    }
}

// ---------------------------------------------------------------------------
// K3: conv 3x3 (9->16, pad 1), no activation.
// p2[512,9,32,32] -> h[512,16*32*32] (flat layout == reshape(B,-1))
// One image per block. Each work item computes a full 32-wide output row via
// a 34-entry register row buffer: weights reused 32x, LDS reads ~0.35/FMA,
// and the 32 consecutive outputs coalesce into b128 stores.
// ---------------------------------------------------------------------------
__global__ __launch_bounds__(256) void conv3_flat(
    const float* __restrict__ p2, const float* __restrict__ k3,
    float* __restrict__ h)
{
    __shared__ float sIn[9 * 32 * 32];
    const int b   = blockIdx.x;
    const int tid = threadIdx.x;

    for (int idx = tid; idx < 9 * 32 * 32; idx += 256)
        sIn[idx] = p2[b * 9 * 32 * 32 + idx];
    __syncthreads();

    for (int item = tid; item < 16 * 32; item += 256) {   // (o, y) pairs
        const int o = item >> 5;
        const int y = item & 31;
        float acc[32];
        #pragma unroll
        for (int i = 0; i < 32; ++i) acc[i] = 0.0f;

        for (int c = 0; c < 9; ++c) {
            #pragma unroll
            for (int ki = 0; ki < 3; ++ki) {
                const int iy = y + ki - 1;
                if ((unsigned)iy < 32u) {
                    float rowBuf[34];                 // ix = -1 .. 32
                    rowBuf[0] = 0.0f; rowBuf[33] = 0.0f;
                    const float* rp = &sIn[c * 1024 + iy * 32];
                    #pragma unroll
                    for (int i = 0; i < 32; ++i) rowBuf[i + 1] = rp[i];
                    #pragma unroll
                    for (int kj = 0; kj < 3; ++kj) {
                        const float w = k3[(c * 9 + ki * 3 + kj) * 16 + o];
                        #pragma unroll
                        for (int xc = 0; xc < 32; ++xc)
                            acc[xc] = fmaf(rowBuf[xc + kj], w, acc[xc]);
                    }
                }
            }
        }
        float* op = &h[(size_t)b * 16384 + (size_t)o * 1024 + (size_t)y * 32];
        #pragma unroll
        for (int xc = 0; xc < 32; ++xc) op[xc] = acc[xc];
    }
}

// ---------------------------------------------------------------------------
// K4: logits = h @ fc_w^T + fc_b, softmax.  M=512, K=16384, N=6 (pad to 16).
// V_WMMA_F32_16X16X4_F32; 4 waves split K, LDS reduction, fused softmax.
// (Unchanged from round 1 — verified to lower to v_wmma_f32_16x16x4_f32.)
// ---------------------------------------------------------------------------
__global__ __launch_bounds__(128) void fc_softmax_wmma(
    const float* __restrict__ h, const float* __restrict__ fc_w,
    const float* __restrict__ fc_b, float* __restrict__ out)
{
    __shared__ float sC[4 * 256];
    __shared__ float sLog[16][8];

    const int m0   = blockIdx.x * 16;
    const int tid  = threadIdx.x;
    const int wave = tid >> 5;
    const int lane = tid & 31;

    const int row = lane & 15;
    const int kk  = (lane >> 4) * 2;
    const int n   = lane & 15;
    const float bMask = (n < 6) ? 1.0f : 0.0f;
    const int nClamped = (n < 6) ? n : 0;

    const float* aBase = h + (size_t)(m0 + row) * 16384 + kk;
    const float* bBase = fc_w + (size_t)nClamped * 16384 + kk;

    const int kBeg = wave * 4096;
    const int kEnd = kBeg + 4096;

    v8f c = {0.f, 0.f, 0.f, 0.f, 0.f, 0.f, 0.f, 0.f};
    for (int k = kBeg; k < kEnd; k += 4) {
        v2f a  = *(const v2f*)(aBase + k);
        v2f bb = *(const v2f*)(bBase + k);
        bb *= bMask;
        c = __builtin_amdgcn_wmma_f32_16x16x4_f32(
                false, a, false, bb, (short)0, c, false, false);
    }

    #pragma unroll
    for (int i = 0; i < 8; ++i)
        sC[wave * 256 + i * 32 + lane] = c[i];
    __syncthreads();

    for (int t = tid; t < 96; t += 128) {
        int m = t / 6, nn = t - m * 6;
        int i = m & 7;
        int l = (m < 8) ? nn : (16 + nn);
        float s = 0.0f;
        #pragma unroll
        for (int w = 0; w < 4; ++w) s += sC[w * 256 + i * 32 + l];
        sLog[m][nn] = s + fc_b[nn];
    }
    __syncthreads();

    if (tid < 16) {
        float mx = -INFINITY;
        #pragma unroll
        for (int nn = 0; nn < 6; ++nn) mx = fmaxf(mx, sLog[tid][nn]);
        float e[6], den = 0.0f;
        #pragma unroll
        for (int nn = 0; nn < 6; ++nn) { e[nn] = expf(sLog[tid][nn] - mx); den += e[nn]; }
        float inv = 1.0f / den;
        #pragma unroll
        for (int nn = 0; nn < 6; ++nn)
            out[(size_t)(m0 + tid) * 6 + nn] = e[nn] * inv;
    }
}

// ---------------------------------------------------------------------------
extern "C" void kernel_launch(void* const* d_in, const int* in_sizes, int n_in,
                              void* d_out, int out_size, void* d_ws, size_t ws_size,
                              hipStream_t stream) {
    (void)in_sizes; (void)n_in; (void)out_size; (void)ws_size;
    const float* x     = (const float*)d_in[0];
    const float* k1    = (const float*)d_in[1];
    const float* gamma = (const float*)d_in[2];
    const float* beta  = (const float*)d_in[3];
    const float* k2    = (const float*)d_in[4];
    const float* k3    = (const float*)d_in[5];
    const float* fc_w  = (const float*)d_in[6];
    const float* fc_b  = (const float*)d_in[7];
    float* outp = (float*)d_out;

    float* p1 = (float*)d_ws;                          // 512*5*63*63  (40.6 MB)
    float* p2 = p1 + (size_t)512 * 5 * 63 * 63;        // 512*9*32*32  (18.9 MB)
    float* h  = p2 + (size_t)512 * 9 * 32 * 32;        // 512*16384    (33.6 MB)

    conv1_bn_pool<<<dim3(16, 512), 256, 0, stream>>>(x, k1, gamma, beta, p1);
    conv2_pool  <<<dim3(4, 512),  256, 0, stream>>>(p1, k2, p2);
    conv3_flat  <<<dim3(512),     256, 0, stream>>>(p2, k3, h);
    fc_softmax_wmma<<<dim3(32),   128, 0, stream>>>(h, fc_w, fc_b, outp);
}